// YoloWrapper_44607530336613
// MI455X (gfx1250) — compile-verified
//
#include <hip/hip_runtime.h>
#include <stdint.h>

// ---------------- problem constants ----------------
#define BATCH   32
#define NANCH   25200
#define NCLS    80
#define ROWDW   85          // 4 box + 1 obj + 80 cls
#define TOPK    1000
#define PADK    1024
#define MAXDET  300
#define CONF_T  0.25f
#define IOU_T   0.45f
#define MAX_WH  4096.0f
#define EPS_F   1e-7f

// kernel-1 tiling: 7 slices of 3600 anchors, chunks of 128 rows (tail 16)
#define SLICES  7
#define SLANCH  3600
#define CHROWS  128
#define NCHUNK  29          // 28 * 128 + 16 = 3600
#define CHDW    (CHROWS * ROWDW)   // 10880 dwords = 43520 B (16B aligned)
#define TAILROWS (SLANCH - (NCHUNK - 1) * CHROWS)   // 16
#define TAILDW  (TAILROWS * ROWDW)                  // 1360 dwords

typedef uint32_t v4u __attribute__((ext_vector_type(4)));
typedef uint32_t v8u __attribute__((ext_vector_type(8)));

// ---------------- order-preserving float<->uint ----------------
__device__ __forceinline__ uint32_t ordf(float x) {
  uint32_t u = __float_as_uint(x);
  return (u & 0x80000000u) ? ~u : (u | 0x80000000u);
}
__device__ __forceinline__ float unordf(uint32_t u) {
  return (u & 0x80000000u) ? __uint_as_float(u & 0x7fffffffu)
                           : __uint_as_float(~u);
}

// =====================================================================
// Kernel 1: stream pred through LDS. Full 43.5 KB chunks are moved by the
// Tensor Data Mover (one tensor_load_to_lds per chunk, issued by wave 0,
// tracked on TENSORcnt); the 5 KB tail chunk uses per-lane
// global_load_async_to_lds_b128 (ASYNCcnt). Double-buffered so the DMA of
// chunk c+1 overlaps the argmax compute on chunk c.
// grid = (SLICES, BATCH), 1024 threads, dyn LDS = 2 * CHDW * 4 bytes.
// =====================================================================
__global__ __launch_bounds__(1024)
void yolo_score_kernel(const float* __restrict__ pred,
                       uint32_t* __restrict__ rawkeys)
{
  extern __shared__ uint32_t stg[];             // double buffer, 2 * 10880 dw
  const int slice = blockIdx.x;
  const int b     = blockIdx.y;
  const int tid   = threadIdx.x;
  const size_t sbase = ((size_t)b * NANCH + (size_t)slice * SLANCH) * ROWDW;
  const uint32_t lds_base = (uint32_t)(size_t)stg;   // LDS byte offset of dyn smem

  // ---- TDM descriptor issue: one instruction moves a whole chunk ----
  auto issue_tdm = [&](int c, int buf) {
    const uint32_t ndw = (uint32_t)CHDW;            // full chunks only
    const uint64_t ga  = (uint64_t)(const char*)(pred + sbase + (size_t)c * CHDW);
    const uint32_t l0  = lds_base + (uint32_t)buf * (uint32_t)(CHDW * 4);
    v4u g0; v8u g1;
    // D# group 0: count=1 (valid user descriptor), lds_addr, global_addr, type=2
    g0[0] = 1u;
    g0[1] = l0;
    g0[2] = (uint32_t)ga;
    g0[3] = (uint32_t)(ga >> 32) | (2u << 30);
    // D# group 1: data_size=2 (4 B elems); 1-D tile flattened as ndw x 1
    g1[0] = 0x00020000u;                    // no multicast/pad/iterate/barrier
    g1[1] = (ndw & 0xFFFFu) << 16;          // tensor_dim0[15:0]  @ bits[79:64... (63:48)]
    g1[2] = (ndw >> 16) | (1u << 16);       // tensor_dim0[31:16], tensor_dim1=1
    g1[3] = (ndw & 0xFFFFu) << 16;          // tensor_dim1[31:16]=0, tile_dim0=ndw
    g1[4] = 1u;                             // tile_dim1=1, tile_dim2=0
    g1[5] = ndw;                            // tensor_dim0_stride[31:0]
    g1[6] = 0u;                             // stride0[47:32]=0, stride1[15:0]=0
    g1[7] = 0u;                             // stride1[47:16]=0
    asm volatile("tensor_load_to_lds %0, %1" :: "s"(g0), "s"(g1) : "memory");
  };

  // ---- per-lane async fallback for the small tail chunk ----
  auto issue_async_tail = [&](int buf) {
    const int n128 = TAILDW >> 2;                   // 340 b128 transfers
    const char* g  = (const char*)(pred + sbase + (size_t)(NCHUNK - 1) * CHDW);
    const uint32_t l0 = lds_base + (uint32_t)buf * (uint32_t)(CHDW * 4);
    for (int i = tid; i < n128; i += 1024) {
      uint32_t lo = l0 + (uint32_t)i * 16u;
      const void* ga = g + (size_t)i * 16;
      asm volatile("global_load_async_to_lds_b128 %0, %1, off"
                   :: "v"(lo), "v"(ga) : "memory");
    }
  };

  if (tid < 32) issue_tdm(0, 0);                    // prefetch chunk 0 (wave 0)

  for (int c = 0; c < NCHUNK; ++c) {
    // wait for chunk c (only chunk c is outstanding at this point)
    asm volatile("s_wait_asynccnt 0" ::: "memory");
    if (tid < 32) __builtin_amdgcn_s_wait_tensorcnt(0);
    __syncthreads();                                // chunk c visible to all
    // launch chunk c+1 into the other buffer (overlaps compute below)
    if (c + 1 < NCHUNK) {
      if (c + 1 == NCHUNK - 1) issue_async_tail((c + 1) & 1);
      else if (tid < 32)       issue_tdm(c + 1, (c + 1) & 1);
    }

    const int rows = (c == NCHUNK - 1) ? TAILROWS : CHROWS;
    const uint32_t* base = stg + (c & 1) * CHDW;
    const int group = tid >> 3;          // 8 lanes per anchor row
    const int sub   = tid & 7;
    if (group < rows) {
      const uint32_t* r = base + group * ROWDW;
      float mv = -1.0f; int mi = 0;
#pragma unroll
      for (int k = 0; k < 10; ++k) {
        int cc = sub + k * 8;            // covers classes 0..79
        float v = __uint_as_float(r[5 + cc]);
        if (v > mv) { mv = v; mi = cc; }
      }
      for (int off = 4; off > 0; off >>= 1) {          // reduce 8 lanes (wave32)
        float ov = __shfl_down(mv, off, 8);
        int   oi = __shfl_down(mi, off, 8);
        if (ov > mv || (ov == mv && oi < mi)) { mv = ov; mi = oi; }
      }
      if (sub == 0) {
        float obj  = __uint_as_float(r[4]);
        float conf = obj * mv;
        rawkeys[(size_t)b * NANCH + slice * SLANCH + c * CHROWS + group] = ordf(conf);
      }
    }
    __syncthreads();                                   // buffer reuse fence
  }
}

// =====================================================================
// Kernel 2: per-batch top-1000 (radix select + bitonic sort),
// bitmask NMS (single-wave serial scan), emit top-300.
// 32 blocks, 1024 threads, ~173 KB dynamic LDS.
// =====================================================================
#define OFF_MAT   0                       // u32[1024*32]  = 131072 B
#define OFF_SB    131072                  // u64[1024]     =   8192 B
#define OFF_BOX   139264                  // 4 * f32[1024] =  16384 B
#define OFF_AUX   155648                  // 4 * f32[1024] =  16384 B
#define OFF_HIST  172032                  // u32[256]      =   1024 B
#define OFF_MISC  173056                  // gmax + scalars + keep words
#define SMEM2_BYTES (OFF_MISC + 256)

__global__ __launch_bounds__(1024)
void yolo_nms_kernel(const float* __restrict__ pred,
                     const uint32_t* __restrict__ rawkeys,
                     float* __restrict__ out)
{
  extern __shared__ unsigned char smem[];
  uint32_t*           mat  = (uint32_t*)(smem + OFF_MAT);
  unsigned long long* sb   = (unsigned long long*)(smem + OFF_SB);
  float* bx1  = (float*)(smem + OFF_BOX);
  float* by1  = bx1 + PADK;
  float* bx2  = by1 + PADK;
  float* by2  = bx2 + PADK;
  float* offv = (float*)(smem + OFF_AUX);
  float* area = offv + PADK;
  float* scv  = area + PADK;
  float* clsv = scv  + PADK;
  uint32_t* hist = (uint32_t*)(smem + OFF_HIST);
  unsigned long long* gmax = (unsigned long long*)(smem + OFF_MISC);
  uint32_t* s_pref = (uint32_t*)(smem + OFF_MISC + 8);
  int*      s_need = (int*)     (smem + OFF_MISC + 12);
  uint32_t* c_gt   = (uint32_t*)(smem + OFF_MISC + 16);
  uint32_t* c_eq   = (uint32_t*)(smem + OFF_MISC + 20);
  uint32_t* keepw  = (uint32_t*)(smem + OFF_MISC + 24);   // 32 words

  const int b   = blockIdx.x;
  const int tid = threadIdx.x;
  const uint32_t* keys = rawkeys + (size_t)b * NANCH;
  const uint32_t ORD_T  = ordf(CONF_T);
  const uint32_t ORD_N1 = ordf(-1.0f);

  // ---- 1. forced detection: argmax conf (min index on ties) ----
  if (tid == 0) *gmax = 0ull;
  __syncthreads();
  {
    unsigned long long lm = 0ull;
    for (int i = tid; i < NANCH; i += 1024) {
      unsigned long long p = ((unsigned long long)keys[i] << 32) |
                             (uint32_t)(~(uint32_t)i);
      if (p > lm) lm = p;
    }
    atomicMax(gmax, lm);
  }
  __syncthreads();
  const uint32_t forced_idx = ~(uint32_t)(*gmax);

  auto keff = [&](int i) -> uint32_t {
    uint32_t k = keys[i];
    if ((uint32_t)i == forced_idx) return k;     // forced valid
    return (k > ORD_T) ? k : ORD_N1;             // score = conf or -1.0
  };

  // ---- 2. radix select: value of the 1000th largest key ----
  if (tid == 0) { *s_pref = 0u; *s_need = TOPK; }
  __syncthreads();
  for (int p = 3; p >= 0; --p) {
    for (int h = tid; h < 256; h += 1024) hist[h] = 0u;
    __syncthreads();
    const uint32_t maskA = (p == 3) ? 0u : (0xFFFFFFFFu << ((p + 1) * 8));
    const uint32_t pf = *s_pref;
    for (int i = tid; i < NANCH; i += 1024) {
      uint32_t k = keff(i);
      if ((k & maskA) == (pf & maskA))
        atomicAdd(&hist[(k >> (p * 8)) & 0xFFu], 1u);
    }
    __syncthreads();
    if (tid == 0) {
      int nd = *s_need; int chosen = 0;
      for (int bb = 255; bb >= 0; --bb) {
        int c = (int)hist[bb];
        if (nd <= c) { chosen = bb; break; }
        nd -= c;
      }
      *s_pref = pf | ((uint32_t)chosen << (p * 8));
      *s_need = nd;
    }
    __syncthreads();
  }
  const uint32_t T   = *s_pref;
  const int      rem = *s_need;
  const int      G   = TOPK - rem;

  // ---- 3. collect the 1000 selected (sort canonicalizes order) ----
  if (tid == 0) { *c_gt = 0u; *c_eq = 0u; }
  __syncthreads();
  for (int i = tid; i < NANCH; i += 1024) {
    uint32_t k = keff(i);
    if (k > T) {
      uint32_t pidx = atomicAdd(c_gt, 1u);
      sb[pidx] = ((unsigned long long)(~k) << 32) | (uint32_t)i;
    } else if (k == T) {
      uint32_t e = atomicAdd(c_eq, 1u);
      if ((int)e < rem)
        sb[G + e] = ((unsigned long long)(~k) << 32) | (uint32_t)i;
    }
  }
  if (tid < PADK - TOPK) sb[TOPK + tid] = ~0ull;   // pad sorts last
  __syncthreads();

  // ---- 4. bitonic sort ascending on (~key, idx): desc score, stable idx ----
  for (unsigned kk = 2; kk <= PADK; kk <<= 1) {
    for (unsigned j = kk >> 1; j > 0; j >>= 1) {
      unsigned i = (unsigned)tid, ixj = i ^ j;
      if (ixj > i) {
        unsigned long long a = sb[i], c = sb[ixj];
        bool up = ((i & kk) == 0);
        if (up ? (a > c) : (a < c)) { sb[i] = c; sb[ixj] = a; }
      }
      __syncthreads();
    }
  }

  // ---- 5. gather rows: boxes, class, score ----
  {
    unsigned long long it = sb[tid];
    if (tid < TOPK) {
      uint32_t idx = (uint32_t)it;
      uint32_t key = ~(uint32_t)(it >> 32);
      float sc = unordf(key);
      const float* r = pred + ((size_t)b * NANCH + idx) * ROWDW;
      float cx = r[0], cy = r[1], w = r[2], h = r[3];
      float best = r[5]; int bi = 0;
      for (int c = 1; c < NCLS; ++c) {
        float v = r[5 + c];
        if (v > best) { best = v; bi = c; }
      }
      float x1 = cx - w * 0.5f, y1 = cy - h * 0.5f;
      float x2 = cx + w * 0.5f, y2 = cy + h * 0.5f;
      bx1[tid] = x1; by1[tid] = y1; bx2[tid] = x2; by2[tid] = y2;
      offv[tid] = (float)bi * MAX_WH;
      area[tid] = (x2 - x1) * (y2 - y1);
      scv[tid]  = sc;
      clsv[tid] = (float)bi;
    } else {
      bx1[tid] = by1[tid] = bx2[tid] = by2[tid] = 0.0f;
      offv[tid] = 0.0f; area[tid] = 0.0f; scv[tid] = -1.0f; clsv[tid] = 0.0f;
    }
  }
  __syncthreads();

  // ---- 6. suppression bit-matrix: mat[i] marks j>i with IoU>thresh ----
  {
    const int i = tid;
    const float o = offv[i];
    const float ix1 = bx1[i] + o, iy1 = by1[i] + o;
    const float ix2 = bx2[i] + o, iy2 = by2[i] + o;
    const float ai  = area[i];
    for (int w = 0; w < 32; ++w) {
      uint32_t bits = 0u;
      for (int l = 0; l < 32; ++l) {
        int j = w * 32 + l;
        float oj = offv[j];
        float iw = fminf(ix2, bx2[j] + oj) - fmaxf(ix1, bx1[j] + oj);
        float ih = fminf(iy2, by2[j] + oj) - fmaxf(iy1, by1[j] + oj);
        iw = fmaxf(iw, 0.0f); ih = fmaxf(ih, 0.0f);
        float inter = iw * ih;
        float iou = inter / (ai + area[j] - inter + EPS_F);
        bits |= ((uint32_t)((iou > IOU_T) && (j > i))) << l;
      }
      mat[i * 32 + w] = bits;
    }
  }
  __syncthreads();

  // ---- 7. serial greedy scan: one wave32, 32-bit keep word per lane ----
  if (tid < 32) {
    uint32_t kw = 0u;
    for (int m = 0; m < 32; ++m)
      kw |= ((uint32_t)(scv[tid * 32 + m] > 0.0f)) << m;
    for (int i = 0; i < TOPK; ++i) {
      uint32_t ow = __shfl(kw, i >> 5, 32);
      if ((ow >> (i & 31)) & 1u)
        kw &= ~mat[i * 32 + tid];
    }
    keepw[tid] = kw;
  }
  __syncthreads();

  // ---- 8. emit first <=300 kept (already sorted desc) ----
  if (tid == 0) {
    float* det = out + (size_t)b * MAXDET * 6;
    float* msk = out + (size_t)BATCH * MAXDET * 6 + (size_t)b * MAXDET;
    int cnt = 0;
    for (int j = 0; j < TOPK && cnt < MAXDET; ++j) {
      if ((keepw[j >> 5] >> (j & 31)) & 1u) {
        det[cnt * 6 + 0] = bx1[j];
        det[cnt * 6 + 1] = by1[j];
        det[cnt * 6 + 2] = bx2[j];
        det[cnt * 6 + 3] = by2[j];
        det[cnt * 6 + 4] = scv[j];
        det[cnt * 6 + 5] = clsv[j];
        msk[cnt] = 1.0f;
        ++cnt;
      }
    }
    for (; cnt < MAXDET; ++cnt) {
      det[cnt * 6 + 0] = 0.0f; det[cnt * 6 + 1] = 0.0f;
      det[cnt * 6 + 2] = 0.0f; det[cnt * 6 + 3] = 0.0f;
      det[cnt * 6 + 4] = 0.0f; det[cnt * 6 + 5] = 0.0f;
      msk[cnt] = 0.0f;
    }
  }
}

// =====================================================================
extern "C" void kernel_launch(void* const* d_in, const int* in_sizes, int n_in,
                              void* d_out, int out_size, void* d_ws, size_t ws_size,
                              hipStream_t stream) {
  (void)in_sizes; (void)n_in; (void)out_size; (void)ws_size;
  const float* pred  = (const float*)d_in[0];
  uint32_t* rawkeys  = (uint32_t*)d_ws;          // BATCH*NANCH u32 = 3.2 MB
  float* out         = (float*)d_out;

  dim3 g1(SLICES, BATCH);
  yolo_score_kernel<<<g1, 1024, 2 * CHDW * sizeof(uint32_t), stream>>>(pred, rawkeys);
  yolo_nms_kernel<<<BATCH, 1024, SMEM2_BYTES, stream>>>(pred, rawkeys, out);
}